// GeneratePrediction_3169685864972
// MI455X (gfx1250) — compile-verified
//
// CDNA5 / gfx1250 implementation of GeneratePrediction (top-k + beam search).
// Memory-bound streaming problem (~393 MB read -> ~17us roofline @ 23.3 TB/s).
// Uses the gfx1250 async global->LDS DMA path for the streaming scans.

#include <hip/hip_runtime.h>
#include <stdint.h>

#define TOPK_K   10
#define B_       4096
#define S_       8
#define VSC_     8000
#define VCO_     2000

#define TILE_F   2048     // floats staged per tile (8 KB)
#define TKWAVES  4        // waves (rows) per block in the top-k kernels
#define TKBLK    (TKWAVES * 32)
#define BEAM_BLK 64

#if __has_builtin(__builtin_amdgcn_global_load_async_to_lds_b128)
#define HAVE_ASYNC_LDS 1
#else
#define HAVE_ASYNC_LDS 0
#endif

// The builtin's params are pointers to int4 (vector_size(16)) in AS1 / AS3.
typedef int v4i __attribute__((vector_size(16)));
#define TO_GV4(p) ((__attribute__((address_space(1))) v4i*)(uintptr_t)(p))
#define TO_LV4(p) ((__attribute__((address_space(3))) v4i*)(uint32_t)(uintptr_t)(p))

// Monotonic float<->uint key so (key<<32 | ~idx) orders by value desc, idx asc.
__device__ __forceinline__ unsigned f2key(float f) {
    unsigned b = __float_as_uint(f);
    return b ^ ((b & 0x80000000u) ? 0xFFFFFFFFu : 0x80000000u);
}
__device__ __forceinline__ float key2f(unsigned k) {
    unsigned b = k ^ ((k & 0x80000000u) ? 0x80000000u : 0xFFFFFFFFu);
    return __uint_as_float(b);
}

__device__ __forceinline__ unsigned long long wave_max_u64(unsigned long long v) {
#pragma unroll
    for (int off = 16; off > 0; off >>= 1) {
        unsigned long long o = __shfl_xor(v, off, 32);
        if (o > v) v = o;
    }
    return v;
}

// One wave per row: stream N f32 through LDS tiles (async DMA), keep a sorted
// per-lane top-10 in registers, then wave-merge 32x10 candidates via shuffles.
// mode 0: write indices to out_idx. mode 1: write indices + values.
__global__ void __launch_bounds__(TKBLK) topk_rows_kernel(
    const float* __restrict__ src, int N, int mode,
    int* __restrict__ out_idx, float* __restrict__ out_val)
{
    __shared__ float stage[TKWAVES][TILE_F];
    __shared__ unsigned long long cand[TKWAVES][32 * TOPK_K];

    const int wave = threadIdx.x >> 5;
    const int lane = threadIdx.x & 31;
    const long long row = (long long)blockIdx.x * TKWAVES + wave;
    const float* rowp = src + row * (long long)N;
    const int rowBytes = N * 4;

    float tv[TOPK_K];
    int   ti[TOPK_K];
#pragma unroll
    for (int j = 0; j < TOPK_K; ++j) { tv[j] = -__builtin_inff(); ti[j] = 0x7FFFFFFF; }

    const int nTiles = (N + TILE_F - 1) / TILE_F;
#pragma unroll 1
    for (int t = 0; t < nTiles; ++t) {
        const int base = t * TILE_F;
        const int cnt  = min(TILE_F, N - base);
#if HAVE_ASYNC_LDS
        // Async DMA the tile into this wave's LDS buffer: 16 B/lane per issue,
        // 512 B per wave-issue, up to 16 issues in flight (8 KB).
        const int issues = (cnt * 4 + 511) / 512;
#pragma unroll 1
        for (int s = 0; s < issues; ++s) {
            int goff = base * 4 + s * 512 + lane * 16;
            if (goff > rowBytes - 16) goff = rowBytes - 16;   // clamp tail (dupes ignored)
            __builtin_amdgcn_global_load_async_to_lds_b128(
                TO_GV4((const char*)rowp + goff),
                TO_LV4((char*)&stage[wave][0] + s * 512 + lane * 16),
                0, 0);
        }
#if __has_builtin(__builtin_amdgcn_s_wait_asynccnt)
        __builtin_amdgcn_s_wait_asynccnt(0);
#else
        asm volatile("s_wait_asynccnt 0" ::: "memory");
#endif
        asm volatile("" ::: "memory");
#else
        // Fallback: plain wave copy through LDS (same consume path).
#pragma unroll 1
        for (int j = lane; j < cnt; j += 32) stage[wave][j] = rowp[base + j];
#endif
        // Consume: per-lane sorted top-10 (common case: 1 compare/elem).
#pragma unroll 1
        for (int j = lane; j < cnt; j += 32) {
            const float v = stage[wave][j];
            if (v > tv[TOPK_K - 1]) {
                tv[TOPK_K - 1] = v; ti[TOPK_K - 1] = base + j;
#pragma unroll
                for (int q = TOPK_K - 1; q > 0; --q) {
                    if (tv[q] > tv[q - 1]) {
                        float fv = tv[q]; tv[q] = tv[q - 1]; tv[q - 1] = fv;
                        int   iv = ti[q]; ti[q] = ti[q - 1]; ti[q - 1] = iv;
                    }
                }
            }
        }
    }

    // Spill own sorted candidates (packed) to LDS for dynamic head indexing.
#pragma unroll
    for (int j = 0; j < TOPK_K; ++j) {
        unsigned long long pk =
            ((unsigned long long)f2key(tv[j]) << 32) | (unsigned)(~(unsigned)ti[j]);
        cand[wave][lane * TOPK_K + j] = pk;
    }
    asm volatile("" ::: "memory");

    // 10 rounds of wave-wide max over the 32 sorted-list heads.
    int head = 0;
    unsigned long long my  = cand[wave][lane * TOPK_K];
    unsigned long long res = 0;
#pragma unroll 1
    for (int j = 0; j < TOPK_K; ++j) {
        const unsigned long long m = wave_max_u64(my);
        if (lane == j) res = m;
        if (my == m) {
            ++head;
            my = (head < TOPK_K) ? cand[wave][lane * TOPK_K + head] : 0ull;
        }
    }

    if (lane < TOPK_K) {
        const int   gi = (int)(~(unsigned)res);
        out_idx[row * TOPK_K + lane] = gi;
        if (mode == 1) out_val[row * TOPK_K + lane] = key2f((unsigned)(res >> 32));
    }
}

// One thread per batch row: 10-wide beam over 8 steps (state in LDS because it
// needs dynamic indexing). Insertion in flat (r*10+c) order matches lax.top_k
// tie-breaking. log_softmax elided: shift-invariant for all returned indices.
__global__ void __launch_bounds__(BEAM_BLK) beam_kernel(
    const float* __restrict__ wsv, const int* __restrict__ wsi,
    int* __restrict__ out_seq)
{
    __shared__ int   seqbuf[2][BEAM_BLK][TOPK_K][S_];
    __shared__ float scrb[BEAM_BLK][TOPK_K];
    __shared__ float cv[BEAM_BLK][TOPK_K];
    __shared__ int   crc[BEAM_BLK][TOPK_K];

    const int t = threadIdx.x;
    const long long b = (long long)blockIdx.x * BEAM_BLK + t;
    const float* v  = wsv + b * (S_ * TOPK_K);
    const int*   ix = wsi + b * (S_ * TOPK_K);

    for (int j = 0; j < TOPK_K; ++j) {
        scrb[t][j] = v[j];
        for (int s = 0; s < S_; ++s) seqbuf[0][t][j][s] = 0;
        seqbuf[0][t][j][0] = ix[j];
    }

    int cur = 0;
#pragma unroll 1
    for (int i = 1; i < S_; ++i) {
        for (int j = 0; j < TOPK_K; ++j) { cv[t][j] = -__builtin_inff(); crc[t][j] = 0; }
#pragma unroll 1
        for (int r = 0; r < TOPK_K; ++r) {
            const float s0 = scrb[t][r];
#pragma unroll 1
            for (int c = 0; c < TOPK_K; ++c) {
                const float val = s0 + v[i * TOPK_K + c];
                if (val > cv[t][TOPK_K - 1]) {
                    cv[t][TOPK_K - 1] = val; crc[t][TOPK_K - 1] = r * TOPK_K + c;
                    for (int q = TOPK_K - 1; q > 0; --q) {
                        if (cv[t][q] > cv[t][q - 1]) {
                            float fv = cv[t][q]; cv[t][q] = cv[t][q - 1]; cv[t][q - 1] = fv;
                            int   iv = crc[t][q]; crc[t][q] = crc[t][q - 1]; crc[t][q - 1] = iv;
                        }
                    }
                }
            }
        }
        const int nxt = cur ^ 1;
        for (int j = 0; j < TOPK_K; ++j) {
            const int rc = crc[t][j];
            const int r = rc / TOPK_K, c = rc % TOPK_K;
            for (int s = 0; s < i; ++s) seqbuf[nxt][t][j][s] = seqbuf[cur][t][r][s];
            seqbuf[nxt][t][j][i] = ix[i * TOPK_K + c];
            for (int s = i + 1; s < S_; ++s) seqbuf[nxt][t][j][s] = 0;
        }
        for (int j = 0; j < TOPK_K; ++j) scrb[t][j] = cv[t][j];
        cur = nxt;
    }

    for (int j = 0; j < TOPK_K; ++j)
        for (int s = 0; s < S_; ++s)
            out_seq[b * (TOPK_K * S_) + j * S_ + s] = seqbuf[cur][t][j][s];
}

// pred_results = where(struc==0, sc_full, where(struc==1, seq, 0))
__global__ void results_kernel(const int* __restrict__ struc,
                               const int* __restrict__ sc_labels,
                               const int* __restrict__ seq,
                               int* __restrict__ out)
{
    const int gid = blockIdx.x * blockDim.x + threadIdx.x;
    if (gid >= B_ * TOPK_K * S_) return;
    const int b = gid / (TOPK_K * S_);
    const int rem = gid - b * (TOPK_K * S_);
    const int j = rem / S_;
    const int s = rem - j * S_;
    const int st = struc[b];
    int val = 0;
    if (st == 0)      val = (s == 0) ? sc_labels[b * TOPK_K + j] : 0;
    else if (st == 1) val = seq[gid];
    out[gid] = val;
}

extern "C" void kernel_launch(void* const* d_in, const int* in_sizes, int n_in,
                              void* d_out, int out_size, void* d_ws, size_t ws_size,
                              hipStream_t stream) {
    (void)in_sizes; (void)n_in; (void)out_size; (void)ws_size;

    const int*   struc        = (const int*)d_in[0];     // (B,)
    const float* sc_logits    = (const float*)d_in[1];   // (B, 8000)
    const float* compo_logits = (const float*)d_in[2];   // (B, 8, 2000)

    int* out0 = (int*)d_out;                    // pred_sc_labels   (B,10)
    int* out1 = out0 + B_ * TOPK_K;             // pred_lr_compo_seq(B,10,8)
    int* out2 = out1 + B_ * TOPK_K * S_;        // pred_results     (B,10,8)

    float* ws_f = (float*)d_ws;                                        // (B*8*10) vals
    int*   ws_i = (int*)((char*)d_ws + (size_t)B_ * S_ * TOPK_K * 4);  // (B*8*10) idx

    // 1) top-10 indices of sc_logits (4096 rows x 8000).
    topk_rows_kernel<<<B_ / TKWAVES, TKBLK, 0, stream>>>(
        sc_logits, VSC_, 0, out0, nullptr);

    // 2) top-10 (val,idx) of compo logits (32768 rows x 2000).
    topk_rows_kernel<<<(B_ * S_) / TKWAVES, TKBLK, 0, stream>>>(
        compo_logits, VCO_, 1, ws_i, ws_f);

    // 3) beam search over the 8 steps.
    beam_kernel<<<B_ / BEAM_BLK, BEAM_BLK, 0, stream>>>(ws_f, ws_i, out1);

    // 4) struc-based select.
    results_kernel<<<(B_ * TOPK_K * S_ + 255) / 256, 256, 0, stream>>>(
        struc, out0, out1, out2);
}